// GraphSAGE_77154792506118
// MI455X (gfx1250) — compile-verified
//
#include <hip/hip_runtime.h>
#include <hip/hip_bf16.h>

typedef __attribute__((ext_vector_type(2))) float v2f;
typedef __attribute__((ext_vector_type(8))) float v8f;

#define K_DIM 128
#define N_NODES_C 40000

// ---------------- utility kernels ----------------

__global__ void zero_f32(float* __restrict__ p, int n) {
    for (int i = blockIdx.x * blockDim.x + threadIdx.x; i < n;
         i += gridDim.x * blockDim.x)
        p[i] = 0.0f;
}

// pad [rows x K] row-major weight into [128 x K], zero-filling extra rows
__global__ void pad_weights(const float* __restrict__ Wsrc,
                            float* __restrict__ Wdst, int rows) {
    int i = blockIdx.x * blockDim.x + threadIdx.x;
    if (i >= 128 * K_DIM) return;
    int r = i / K_DIM;
    Wdst[i] = (r < rows) ? Wsrc[i] : 0.0f;
}

__global__ void combine_bias(const float* __restrict__ bl,
                             const float* __restrict__ br,
                             float* __restrict__ out, int n, int npad) {
    int i = blockIdx.x * blockDim.x + threadIdx.x;
    if (i < npad) out[i] = (i < n) ? (bl[i] + br[i]) : 0.0f;
}

// ---------------- edge phase ----------------

__global__ void count_kernel(const int* __restrict__ dst,
                             float* __restrict__ cnt, int nEdges) {
    int e = blockIdx.x * blockDim.x + threadIdx.x;
    if (e < nEdges) unsafeAtomicAdd(&cnt[dst[e]], 1.0f);
}

// one wave (32 lanes) per edge; each lane moves 4 contiguous channels
__global__ void scatter_kernel(const float* __restrict__ feat,
                               const int* __restrict__ src,
                               const int* __restrict__ dst,
                               float* __restrict__ agg, int nEdges) {
    int e = blockIdx.x * (blockDim.x >> 5) + (threadIdx.x >> 5);
    if (e >= nEdges) return;
    int lane = threadIdx.x & 31;
    int s = src[e];
    int d = dst[e];
    const float4 v =
        *(const float4*)(feat + (size_t)s * K_DIM + lane * 4);
    float* o = agg + (size_t)d * K_DIM + lane * 4;
    unsafeAtomicAdd(o + 0, v.x);
    unsafeAtomicAdd(o + 1, v.y);
    unsafeAtomicAdd(o + 2, v.z);
    unsafeAtomicAdd(o + 3, v.w);
}

// ---------------- fused SAGE GEMM (WMMA fp32 16x16x4) ----------------
// out[m,n] = act( sum_k (agg[m,k]/max(cnt[m],1))*Wl[n,k]
//               + sum_k  hin[m,k]*Wr[n,k]  + bias[n] )
// One wave computes one 16x16 output tile; 8 waves/block cover all 8
// N-tiles of a 16-row node slab.  A-tile load: lane&15 = row, lane>>4
// selects the K pair (matches ISA 32-bit 16x4 A layout); B-tile load is
// the symmetric pattern on the [N x K] row-major weight.
template <bool RELU>
__global__ void sage_gemm(const float* __restrict__ agg,
                          const float* __restrict__ cnt,
                          const float* __restrict__ hin,
                          const float* __restrict__ Wl,   // [128 x 128] (N,K)
                          const float* __restrict__ Wr,   // [128 x 128] (N,K)
                          const float* __restrict__ bias, // [128]
                          float* __restrict__ out, int M, int Nout,
                          int ldOut) {
    const int lane = threadIdx.x & 31;
    const int wave = threadIdx.x >> 5;  // 0..7 -> N tile
    const int m0 = blockIdx.x * 16;
    const int n0 = wave * 16;

    const int half = lane >> 4;   // 0: K+{0,1}, 1: K+{2,3}
    const int l15 = lane & 15;
    const int rowA = m0 + l15;    // node row for A loads
    const int rowB = n0 + l15;    // output-channel row of W for B loads

    const float rs = 1.0f / fmaxf(cnt[rowA], 1.0f);

    const float* aBase = agg + (size_t)rowA * K_DIM + half * 2;
    const float* hBase = hin + (size_t)rowA * K_DIM + half * 2;
    const float* wlBase = Wl + (size_t)rowB * K_DIM + half * 2;
    const float* wrBase = Wr + (size_t)rowB * K_DIM + half * 2;

    v8f c = {};

#pragma unroll
    for (int k = 0; k < K_DIM; k += 4) {
        v2f a = *(const v2f*)(aBase + k);
        a *= rs;  // fold mean-divide into the A tile
        v2f b = *(const v2f*)(wlBase + k);
        c = __builtin_amdgcn_wmma_f32_16x16x4_f32(false, a, false, b,
                                                  (short)0, c, false, false);
    }
#pragma unroll
    for (int k = 0; k < K_DIM; k += 4) {
        v2f a = *(const v2f*)(hBase + k);
        v2f b = *(const v2f*)(wrBase + k);
        c = __builtin_amdgcn_wmma_f32_16x16x4_f32(false, a, false, b,
                                                  (short)0, c, false, false);
    }

    // C/D layout: VGPR r -> row m0+r (lanes 0-15) or m0+r+8 (lanes 16-31),
    // col n0 + (lane&15)
    const int n = n0 + l15;
#pragma unroll
    for (int r = 0; r < 8; ++r) {
        int m = m0 + r + 8 * half;
        if (m < M && n < Nout) {
            float v = c[r] + bias[n];
            if (RELU) v = fmaxf(v, 0.0f);
            out[(size_t)m * ldOut + n] = v;
        }
    }
}

// ---------------- launch ----------------

extern "C" void kernel_launch(void* const* d_in, const int* in_sizes, int n_in,
                              void* d_out, int out_size, void* d_ws,
                              size_t ws_size, hipStream_t stream) {
    const float* x = (const float*)d_in[0];
    const int* ei = (const int*)d_in[1];
    const float* W1l = (const float*)d_in[2];
    const float* b1l = (const float*)d_in[3];
    const float* W1r = (const float*)d_in[4];
    const float* b1r = (const float*)d_in[5];
    const float* W2l = (const float*)d_in[6];
    const float* b2l = (const float*)d_in[7];
    const float* W2r = (const float*)d_in[8];
    const float* b2r = (const float*)d_in[9];

    const int M = in_sizes[0] / K_DIM;      // 40000 nodes
    const int E = in_sizes[1] / 2;          // 640000 edges
    const int OUTC = 121;
    const int* src = ei;
    const int* dst = ei + E;

    // workspace carve (floats)
    float* ws = (float*)d_ws;
    float* cnt = ws;                          // 40960
    float* agg = cnt + 40960;                 // M*128
    float* h = agg + (size_t)M * K_DIM;       // M*128
    float* W2lp = h + (size_t)M * K_DIM;      // 128*128
    float* W2rp = W2lp + 128 * K_DIM;         // 128*128
    float* bias1 = W2rp + 128 * K_DIM;        // 128
    float* bias2 = bias1 + 128;               // 128

    const int T = 256;
    const int edgeBlocks = (E + (T / 32) - 1) / (T / 32);
    const int nAgg = M * K_DIM;

    // prep
    zero_f32<<<(M + T - 1) / T, T, 0, stream>>>(cnt, M);
    zero_f32<<<4096, T, 0, stream>>>(agg, nAgg);
    pad_weights<<<(128 * K_DIM + T - 1) / T, T, 0, stream>>>(W2l, W2lp, OUTC);
    pad_weights<<<(128 * K_DIM + T - 1) / T, T, 0, stream>>>(W2r, W2rp, OUTC);
    combine_bias<<<1, 128, 0, stream>>>(b1l, b1r, bias1, 128, 128);
    combine_bias<<<1, 128, 0, stream>>>(b2l, b2r, bias2, OUTC, 128);

    // degree counts (shared by both layers)
    count_kernel<<<(E + T - 1) / T, T, 0, stream>>>(dst, cnt, E);

    // layer 1: scatter x, then fused (mean | GEMM | bias | relu)
    scatter_kernel<<<edgeBlocks, T, 0, stream>>>(x, src, dst, agg, E);
    sage_gemm<true><<<M / 16, T, 0, stream>>>(agg, cnt, x, W1l, W1r, bias1, h,
                                              M, 128, 128);

    // layer 2: re-zero agg, scatter h, fused GEMM into d_out (N=121)
    zero_f32<<<4096, T, 0, stream>>>(agg, nAgg);
    scatter_kernel<<<edgeBlocks, T, 0, stream>>>(h, src, dst, agg, E);
    sage_gemm<false><<<M / 16, T, 0, stream>>>(agg, cnt, h, W2lp, W2rp, bias2,
                                               (float*)d_out, M, OUTC, OUTC);
}